// GroupedLinear_5093831213305
// MI455X (gfx1250) — compile-verified
//
#include <hip/hip_runtime.h>
#include <stdint.h>

typedef __attribute__((ext_vector_type(2))) float v2f;
typedef __attribute__((ext_vector_type(4))) float v4f;
typedef __attribute__((ext_vector_type(8))) float v8f;

#define T_TOK 16384
#define NGRP  8
#define DIN   2048
#define DOUT  8192
#define BM    128
#define BN    128
#define BK    32
#define NKB   (DIN / BK)        // 64 k-blocks
#define LDS_TILE (BM * BK)      // 4096 floats = 16 KB
#define LDS_BUF  (2 * LDS_TILE) // A tile + W tile per buffer

// Async DMA global->LDS via inline asm (ASYNCcnt path). Set to 0 to fall back
// to register-staged global_load_b128 + ds_store_b128.
#define USE_ASYNC_COPY 1

__device__ __forceinline__ void wait_async0() {
#if __has_builtin(__builtin_amdgcn_s_wait_asynccnt)
    __builtin_amdgcn_s_wait_asynccnt(0);
#else
    asm volatile("s_wait_asynccnt 0x0" ::: "memory");
#endif
}

// XOR swizzle of the K-column so that stride-32 rows never collide on the
// 64 LDS banks; multiples of 4 keep both b128 stores and b64 loads aligned.
__device__ __forceinline__ int swz(int row) { return ((row >> 1) & 7) << 2; }

#if USE_ASYNC_COPY
// One b128 async DMA copy per lane: 16 bytes from global address saddr + voff to the LDS address in ldsaddr.
__device__ __forceinline__ void async_b128(uint32_t ldsaddr, uint32_t voff,
                                           const float* saddr) {
    asm volatile("global_load_async_to_lds_b128 %0, %1, %2"
                 :: "v"(ldsaddr), "v"(voff), "s"(saddr)
                 : "memory");
}

// Issue 8 async DMA b128 copies: one (128x32) f32 tile of X and of W -> LDS.
__device__ __forceinline__ void stage_tiles_async(const float* Xk, const float* Wk,
                                                  uint32_t ldsA_byte,
                                                  const uint32_t* goff,
                                                  const uint32_t* loff) {
#pragma unroll
    for (int p = 0; p < 4; ++p) {
        async_b128(ldsA_byte + loff[p],                 goff[p], Xk);
        async_b128(ldsA_byte + loff[p] + LDS_TILE * 4,  goff[p], Wk);
    }
}
#endif

__global__ __launch_bounds__(256)
void grouped_gemm_f32_wmma(const float* __restrict__ X,
                           const long long* __restrict__ offs,
                           const float* __restrict__ W,
                           float* __restrict__ Y)
{
    __shared__ float lds[2 * LDS_BUF];   // 64 KB, double buffered

    const int tid  = threadIdx.x;
    const int lane = tid & 31;
    const int wave = tid >> 5;
    const int wm   = wave >> 1;          // 0..3  (M direction)
    const int wn   = wave & 1;           // 0..1  (N direction)

    const int row_block = blockIdx.y * BM;
    const int col_block = blockIdx.x * BN;

    // Which expert owns this row tile (boundaries are multiples of BM here).
    int g = 0;
#pragma unroll
    for (int i = 0; i < NGRP; ++i)
        g += (row_block >= (int)offs[i]) ? 1 : 0;

    const float* Xb = X + (size_t)row_block * DIN;
    const float* Wb = W + ((size_t)g * DOUT + (size_t)col_block) * DIN;

    // global -> LDS staging geometry: 4 passes of (32 rows x 8 float4)
    const int lr = tid >> 3;             // 0..31
    const int lc = (tid & 7) << 2;       // 0,4,...,28

    // Per-lane byte offsets for the DMA (identical for X and W tiles).
    uint32_t goff[4], loff[4];
#pragma unroll
    for (int p = 0; p < 4; ++p) {
        int r   = p * 32 + lr;
        goff[p] = (uint32_t)((r * DIN + lc) * 4);
        loff[p] = (uint32_t)((r * BK + (lc ^ swz(r))) * 4);
    }
    const uint32_t lds_base = (uint32_t)(uintptr_t)&lds[0];

    // fragment geometry (A 16x4 and B 4x16 f32 layouts are symmetric:
    // lane reads 2 consecutive K at k + 2*(lane>=16) for its M/N row)
    const int half = lane >> 4;          // 0/1
    const int l15  = lane & 15;
    const int klo  = half << 1;          // 0/2

    int abase[2], asw[2];
#pragma unroll
    for (int i = 0; i < 2; ++i) {
        int r = wm * 32 + i * 16 + l15;
        abase[i] = r * BK;
        asw[i]   = swz(r);
    }
    int bbase[4], bsw[4];
#pragma unroll
    for (int j = 0; j < 4; ++j) {
        int r = wn * 64 + j * 16 + l15;
        bbase[j] = r * BK;
        bsw[j]   = swz(r);
    }

    const v8f vzero = {};
    v8f acc[2][4];
#pragma unroll
    for (int i = 0; i < 2; ++i)
#pragma unroll
        for (int j = 0; j < 4; ++j)
            acc[i][j] = vzero;

#if !USE_ASYNC_COPY
    v4f xr[4], wr[4];
#endif

    // ---- prologue: k-block 0 into buffer 0 ----
#if USE_ASYNC_COPY
    stage_tiles_async(Xb, Wb, lds_base, goff, loff);
    wait_async0();
#else
#pragma unroll
    for (int p = 0; p < 4; ++p) {
        int r = p * 32 + lr;
        xr[p] = *(const v4f*)(Xb + (size_t)r * DIN + lc);
        wr[p] = *(const v4f*)(Wb + (size_t)r * DIN + lc);
    }
#pragma unroll
    for (int p = 0; p < 4; ++p) {
        int r  = p * 32 + lr;
        int cs = lc ^ swz(r);
        *(v4f*)&lds[r * BK + cs]            = xr[p];
        *(v4f*)&lds[LDS_TILE + r * BK + cs] = wr[p];
    }
#endif
    __syncthreads();

    for (int kb = 0; kb < NKB; ++kb) {
        const int cur = kb & 1;
        const float* As = &lds[cur * LDS_BUF];
        const float* Bs = As + LDS_TILE;

        // kick off the DMA (or register prefetch) for the next k-block
        if (kb + 1 < NKB) {
            const float* Xk = Xb + (size_t)(kb + 1) * BK;
            const float* Wk = Wb + (size_t)(kb + 1) * BK;
#if USE_ASYNC_COPY
            stage_tiles_async(Xk, Wk,
                              lds_base + (uint32_t)((cur ^ 1) * LDS_BUF * 4),
                              goff, loff);
#else
#pragma unroll
            for (int p = 0; p < 4; ++p) {
                int r = p * 32 + lr;
                xr[p] = *(const v4f*)(Xk + (size_t)r * DIN + lc);
                wr[p] = *(const v4f*)(Wk + (size_t)r * DIN + lc);
            }
#endif
        }

        // 8 K-steps x (2 M x 4 N) = 64 f32 WMMAs, fragments double-buffered
        // so the DS wait for step kk+4 hides behind the 8 WMMAs of step kk.
        v2f af[2][2], bf[2][4];
#pragma unroll
        for (int i = 0; i < 2; ++i)
            af[0][i] = *(const v2f*)&As[abase[i] + (klo ^ asw[i])];
#pragma unroll
        for (int j = 0; j < 4; ++j)
            bf[0][j] = *(const v2f*)&Bs[bbase[j] + (klo ^ bsw[j])];

#pragma unroll
        for (int kk = 0; kk < BK; kk += 4) {
            const int cb = (kk >> 2) & 1;
            const int nb = cb ^ 1;
            if (kk + 4 < BK) {
#pragma unroll
                for (int i = 0; i < 2; ++i)
                    af[nb][i] = *(const v2f*)&As[abase[i] + ((kk + 4 + klo) ^ asw[i])];
#pragma unroll
                for (int j = 0; j < 4; ++j)
                    bf[nb][j] = *(const v2f*)&Bs[bbase[j] + ((kk + 4 + klo) ^ bsw[j])];
            }
#pragma unroll
            for (int i = 0; i < 2; ++i)
#pragma unroll
                for (int j = 0; j < 4; ++j)
                    acc[i][j] = __builtin_amdgcn_wmma_f32_16x16x4_f32(
                        false, af[cb][i], false, bf[cb][j],
                        (short)0, acc[i][j], false, false);
        }

#if USE_ASYNC_COPY
        wait_async0();                  // my DMA into the other buffer is done
#else
        if (kb + 1 < NKB) {
            float* Ad = &lds[(cur ^ 1) * LDS_BUF];
#pragma unroll
            for (int p = 0; p < 4; ++p) {
                int r  = p * 32 + lr;
                int cs = lc ^ swz(r);
                *(v4f*)&Ad[r * BK + cs]            = xr[p];
                *(v4f*)&Ad[LDS_TILE + r * BK + cs] = wr[p];
            }
        }
#endif
        __syncthreads();
    }

    // ---- epilogue: C/D layout -> global (16 consecutive cols per half-wave)
#pragma unroll
    for (int i = 0; i < 2; ++i) {
        int row0 = row_block + wm * 32 + i * 16 + (half << 3);
#pragma unroll
        for (int j = 0; j < 4; ++j) {
            int col = col_block + wn * 64 + j * 16 + l15;
#pragma unroll
            for (int r = 0; r < 8; ++r)
                Y[(size_t)(row0 + r) * DOUT + col] = acc[i][j][r];
        }
    }
}

extern "C" void kernel_launch(void* const* d_in, const int* in_sizes, int n_in,
                              void* d_out, int out_size, void* d_ws, size_t ws_size,
                              hipStream_t stream) {
    const float*     X    = (const float*)d_in[0];
    const long long* offs = (const long long*)d_in[1];   // int64 cumulative sizes
    const float*     W    = (const float*)d_in[2];
    float*           Y    = (float*)d_out;

    dim3 grid(DOUT / BN, T_TOK / BM);   // 64 x 128 workgroups
    grouped_gemm_f32_wmma<<<grid, dim3(256), 0, stream>>>(X, offs, W, Y);
}